// ClassicalHopfield_54417235640621
// MI455X (gfx1250) — compile-verified
//
#include <hip/hip_runtime.h>

typedef float v2f __attribute__((ext_vector_type(2)));
typedef float v8f __attribute__((ext_vector_type(8)));

#define NEURONS 8192
#define BLK 32       // neurons resolved per block-sequential step (one wave)
#define CSTRIDE 36   // coupling-tile row stride: 16B-aligned, bank-spread

// ---------------------------------------------------------------------------
// Phase 1: act = W @ state  (uses W symmetry: act[n] = sum_k s[k] * W[k][n])
// One wave per 16-column tile; K-loop in steps of 4 via V_WMMA_F32_16X16X4_F32.
// A (16x4): row 0 = state segment  -> lane0: {K0,K1}, lane16: {K2,K3}, rest 0.
// B (4x16): VGPR0 = rows {K0 | K2}, VGPR1 = rows {K1 | K3}, cols n0..n0+15.
// D: row M=0 lands in VGPR0 lanes 0..15 -> act[n0 + lane].
// ---------------------------------------------------------------------------
__global__ __launch_bounds__(256)
void hop_act_wmma(const float* __restrict__ W, const float* __restrict__ s,
                  float* __restrict__ act, int N) {
  __shared__ float s_lds[NEURONS];
  for (int j = threadIdx.x; j < N; j += blockDim.x) s_lds[j] = s[j];
  __syncthreads();

  const int lane = threadIdx.x & 31;
  const int wave = threadIdx.x >> 5;
  const int tile = blockIdx.x * (blockDim.x >> 5) + wave;
  const int n0   = tile << 4;
  const int half = lane >> 4;
  const int l16  = lane & 15;
  const size_t Ns = (size_t)N;

  v8f c = {0.f, 0.f, 0.f, 0.f, 0.f, 0.f, 0.f, 0.f};

  for (int k = 0; k < N; k += 4) {
    const int krow = k + (half << 1);
    const float a0 = s_lds[krow];
    const float a1 = s_lds[krow + 1];
    v2f a;
    a.x = (l16 == 0) ? a0 : 0.0f;
    a.y = (l16 == 0) ? a1 : 0.0f;
    const float* bp = W + (size_t)krow * Ns + (size_t)(n0 + l16);
    v2f b;
    b.x = bp[0];
    b.y = bp[Ns];
    c = __builtin_amdgcn_wmma_f32_16x16x4_f32(false, a, false, b, (short)0, c,
                                              false, false);
  }

  if (half == 0) act[n0 + l16] = c[0];
}

// ---------------------------------------------------------------------------
// Phase 2: block-sequential asynchronous sweep, one workgroup (32 waves/WGP).
// Per 32-neuron block (3 barriers):
//   stage:   park current-block coupling element (register) into LDS tile, read
//            stale activations, load next-block indices
//   resolve: wave 0 runs the 32-step sequential sign chain entirely in
//            registers (coupling row preloaded via 8x b128 LDS loads; chain is
//            shfl+fma only), compacts flips via ballot
//   apply:   all 1024 threads stream the flipped W rows (contiguous b128,
//            multiple rows in flight) into the batched rank-k act update
//   ...and the NEXT block's scattered coupling gather is issued right after
//   the stage barrier, so its HBM latency hides behind resolve+apply.
// ---------------------------------------------------------------------------
__global__ __launch_bounds__(1024, 1)
void hop_seq(const float* __restrict__ W, const float* __restrict__ state_in,
             const int* __restrict__ perm, const int* __restrict__ niter,
             const float* __restrict__ act_in, float* __restrict__ out, int N) {
  __shared__ float    act[NEURONS];            // 32 KB
  __shared__ unsigned stbits[NEURONS / 32];    // 1 KB state bitmask (1 = +1)
  __shared__ float    C[BLK * CSTRIDE];        // 4.5 KB coupling tile
  __shared__ int      idxsA[BLK];
  __shared__ int      idxsB[BLK];
  __shared__ float    a_st[BLK];
  __shared__ int      flist[BLK];
  __shared__ float    fdelta[BLK];
  __shared__ int      nflip;

  const int tid = threadIdx.x;
  const size_t Ns = (size_t)N;

  for (int j = tid; j < N; j += blockDim.x) act[j] = act_in[j];
  if (tid < N / 32) {
    unsigned w = 0u;
    for (int j = 0; j < 32; ++j)
      if (state_in[tid * 32 + j] >= 0.0f) w |= (1u << j);
    stbits[tid] = w;
  }

  const int iters = niter[0];
  const int b8 = tid << 3;          // this thread owns act[b8 .. b8+7]
  const int gb = tid >> 5;          // coupling-gather row (wave id)
  const int gc = tid & 31;          // coupling-gather col (lane id)

  // ---- pipeline prologue: indices + coupling gather for block 0 ----------
  if (tid < BLK) idxsA[tid] = perm[tid];
  __syncthreads();
  float creg = W[(size_t)idxsA[gb] * Ns + (size_t)idxsA[gc]];

  int* cur = idxsA;
  int* nxt = idxsB;

  for (int it = 0; it < iters; ++it) {
    for (int t0 = 0; t0 < N; t0 += BLK) {
      // ---- stage ------------------------------------------------------
      C[gb * CSTRIDE + gc] = creg;      // current block's coupling tile
      if (tid < BLK) {
        a_st[tid] = act[cur[tid]];      // stale activations (post prev apply)
        int tn = t0 + BLK;
        if (tn >= N) tn = 0;            // wraps across sweeps; harmless at end
        nxt[tid] = perm[tn + tid];
      }
      __syncthreads();

      // issue NEXT block's scattered coupling gather; latency hides behind
      // the resolve + apply work below.
      creg = W[(size_t)nxt[gb] * Ns + (size_t)nxt[gc]];

      // ---- resolve (wave 0 only, register speed) ----------------------
      if (tid < BLK) {
        const int i = cur[tid];
        float a = a_st[tid];
        const unsigned wrd = stbits[i >> 5];
        const float s_old = ((wrd >> (i & 31)) & 1u) ? 1.0f : -1.0f;

        float cr[BLK];                  // this lane's coupling row
#pragma unroll
        for (int q = 0; q < BLK / 4; ++q) {
          const float4 t4 = *(const float4*)&C[tid * CSTRIDE + 4 * q];
          cr[4 * q + 0] = t4.x; cr[4 * q + 1] = t4.y;
          cr[4 * q + 2] = t4.z; cr[4 * q + 3] = t4.w;
        }

        float d = 0.0f;
#pragma unroll
        for (int bb = 0; bb < BLK; ++bb) {
          const float a_b  = __shfl(a, bb);
          const float so_b = __shfl(s_old, bb);
          const float sn   = (a_b >= 0.0f) ? 1.0f : -1.0f;
          const float dd   = sn - so_b;               // 0 or +/-2
          if (tid == bb) d = dd;
          if (tid > bb)  a += dd * cr[bb];
        }

        const unsigned mask = (unsigned)__ballot(d != 0.0f);
        const int pos = __popc(mask & ((1u << tid) - 1u));
        if (d != 0.0f) {
          flist[pos]  = i;
          fdelta[pos] = d;
          atomicXor(&stbits[i >> 5], 1u << (i & 31));
        }
        if (tid == 0) nflip = __popc(mask);
      }
      __syncthreads();

      // ---- apply batched rank-k update --------------------------------
      const int nf = nflip;
      if (nf > 0) {
        float4 A0 = {0.f, 0.f, 0.f, 0.f};
        float4 A1 = {0.f, 0.f, 0.f, 0.f};
        for (int f = 0; f < nf; ++f) {
          const float dd = fdelta[f];
          const float4* wr =
              (const float4*)(W + (size_t)flist[f] * Ns) + (tid << 1);
          const float4 w0 = wr[0];
          const float4 w1 = wr[1];
          A0.x += dd * w0.x; A0.y += dd * w0.y;
          A0.z += dd * w0.z; A0.w += dd * w0.w;
          A1.x += dd * w1.x; A1.y += dd * w1.y;
          A1.z += dd * w1.z; A1.w += dd * w1.w;
        }
        act[b8 + 0] += A0.x; act[b8 + 1] += A0.y;
        act[b8 + 2] += A0.z; act[b8 + 3] += A0.w;
        act[b8 + 4] += A1.x; act[b8 + 5] += A1.y;
        act[b8 + 6] += A1.z; act[b8 + 7] += A1.w;
      }
      __syncthreads();

      int* tmp = cur; cur = nxt; nxt = tmp;
    }
  }

  for (int j = tid; j < N; j += blockDim.x) {
    const unsigned w = stbits[j >> 5];
    out[j] = ((w >> (j & 31)) & 1u) ? 1.0f : -1.0f;
  }
}

// ---------------------------------------------------------------------------
extern "C" void kernel_launch(void* const* d_in, const int* in_sizes, int n_in,
                              void* d_out, int out_size, void* d_ws, size_t ws_size,
                              hipStream_t stream) {
  const float* W     = (const float*)d_in[0];   // [N, N] f32, symmetric, 0 diag
  const float* state = (const float*)d_in[1];   // [N] f32, +/-1
  const int*   perm  = (const int*)d_in[2];     // [N] i32
  const int*   niter = (const int*)d_in[3];     // [1] i32
  float*       out   = (float*)d_out;           // [N] f32
  float*       act_g = (float*)d_ws;            // [N] f32 scratch

  const int N = in_sizes[1];                    // 8192
  const int tiles = N >> 4;                     // 512 column tiles
  const int wavesPerBlock = 256 / 32;           // 8
  dim3 grid1(tiles / wavesPerBlock);            // 64 blocks * 8 waves = 512

  hop_act_wmma<<<grid1, dim3(256), 0, stream>>>(W, state, act_g, N);
  hop_seq<<<dim3(1), dim3(1024), 0, stream>>>(W, state, perm, niter, act_g, out, N);
}